// RelationalGNN_47897475285054
// MI455X (gfx1250) — compile-verified
//
#include <hip/hip_runtime.h>
#include <hip/hip_bf16.h>
#include <math.h>

#define NN 20000
#define NE 320000
#define BG 16
#define EA (NE + NN)   // augmented edges (self loops appended)
#define CDIV(a,b) (((a)+(b)-1)/(b))

typedef float v2f __attribute__((ext_vector_type(2)));
typedef float v8f __attribute__((ext_vector_type(8)));

// ---- ordered-int mapping so unsigned atomicMax == float max ----
__device__ __forceinline__ unsigned ford(float f) {
    unsigned u = __float_as_uint(f);
    return (u & 0x80000000u) ? ~u : (u | 0x80000000u);
}
__device__ __forceinline__ float funord(unsigned u) {
    u = (u & 0x80000000u) ? (u ^ 0x80000000u) : ~u;
    return __uint_as_float(u);
}
#define ORD_NEG_INF 0x007FFFFFu   // ford(-inf)

// ------------------------------------------------------------------
// fills
// ------------------------------------------------------------------
__global__ void k_fill_f32(float* p, float v, int n) {
    int i = blockIdx.x * blockDim.x + threadIdx.x;
    if (i < n) p[i] = v;
}
__global__ void k_fill_u32(unsigned* p, unsigned v, int n) {
    int i = blockIdx.x * blockDim.x + threadIdx.x;
    if (i < n) p[i] = v;
}
__global__ void k_copy(const float* __restrict__ a, float* __restrict__ b, int n) {
    int i = blockIdx.x * blockDim.x + threadIdx.x;
    if (i < n) b[i] = a[i];
}

// ------------------------------------------------------------------
// Generic fp32 WMMA GEMM: C[m, cofs+n] = act(sum_k A[m,k]*W[n,k] + bias[n])
// One wave per 16-row tile, NT column tiles of 16. M%16==0, K%4==0.
// ------------------------------------------------------------------
template <int NT>
__global__ __launch_bounds__(256) void k_gemm(
    const float* __restrict__ A, const float* __restrict__ W,
    const float* __restrict__ bias, float* __restrict__ C,
    int M, int K, int lda, int ldw, int ldc, int cofs, int relu)
{
    int lane = threadIdx.x & 31;
    int wv   = threadIdx.x >> 5;
    int m0   = (blockIdx.x * 8 + wv) * 16;
    if (m0 >= M) return;
    int col0 = lane & 15;
    int row  = m0 + col0;
    int koff = (lane < 16) ? 0 : 2;

    v8f acc[NT];
#pragma unroll
    for (int i = 0; i < NT; ++i) acc[i] = (v8f){};

    const float* Arow = A + (size_t)row * lda + koff;
    for (int k0 = 0; k0 < K; k0 += 4) {
        v2f a; a.x = Arow[k0]; a.y = Arow[k0 + 1];
#pragma unroll
        for (int nt = 0; nt < NT; ++nt) {
            int wn = nt * 16 + col0;
            v2f b;
            b.x = W[(size_t)wn * ldw + k0 + koff];
            b.y = W[(size_t)wn * ldw + k0 + koff + 1];
            acc[nt] = __builtin_amdgcn_wmma_f32_16x16x4_f32(
                false, a, false, b, (short)0, acc[nt], false, false);
        }
    }
    int rbase = (lane >> 4) * 8;
#pragma unroll
    for (int nt = 0; nt < NT; ++nt) {
        float bv = bias ? bias[nt * 16 + col0] : 0.f;
#pragma unroll
        for (int r = 0; r < 8; ++r) {
            float v = acc[nt][r] + bv;
            if (relu) v = fmaxf(v, 0.f);
            C[(size_t)(m0 + rbase + r) * ldc + cofs + nt * 16 + col0] = v;
        }
    }
}

// ------------------------------------------------------------------
// Node encoder layer 1: Linear(5->64) -> LN -> ReLU
// ------------------------------------------------------------------
__global__ void k_node_enc1(const float* __restrict__ x, const float* __restrict__ w1,
                            const float* __restrict__ b1, const float* __restrict__ g1,
                            const float* __restrict__ bt1, float* __restrict__ t64, int N)
{
    int n = blockIdx.x * blockDim.x + threadIdx.x;
    if (n >= N) return;
    float xv[5];
#pragma unroll
    for (int k = 0; k < 5; ++k) xv[k] = x[n * 5 + k];
    float y[64]; float s = 0.f;
#pragma unroll
    for (int o = 0; o < 64; ++o) {
        float a = b1[o];
#pragma unroll
        for (int k = 0; k < 5; ++k) a += xv[k] * w1[o * 5 + k];
        y[o] = a; s += a;
    }
    float mu = s * (1.f / 64.f), var = 0.f;
#pragma unroll
    for (int o = 0; o < 64; ++o) { float d = y[o] - mu; var += d * d; }
    float rstd = rsqrtf(var * (1.f / 64.f) + 1e-5f);
#pragma unroll
    for (int o = 0; o < 64; ++o) {
        float v = (y[o] - mu) * rstd * g1[o] + bt1[o];
        t64[(size_t)n * 64 + o] = fmaxf(v, 0.f);
    }
}

__global__ void k_type_concat(const float* __restrict__ emb, const int* __restrict__ types,
                              float* __restrict__ h, int N)
{
    int i = blockIdx.x * blockDim.x + threadIdx.x;
    if (i >= N * 32) return;
    int n = i >> 5, c = i & 31;
    h[(size_t)n * 128 + 96 + c] = emb[types[n] * 32 + c];
}

// ------------------------------------------------------------------
// Edge encoder: Linear(2->32) -> LN -> ReLU, plus self-loop mean accum
// ------------------------------------------------------------------
__global__ void k_edge_enc(const float* __restrict__ eattr, const float* __restrict__ w,
                           const float* __restrict__ b, const float* __restrict__ g,
                           const float* __restrict__ bt, const int* __restrict__ tgt,
                           float* __restrict__ eaA, float* __restrict__ cnt, int E)
{
    int e = blockIdx.x * blockDim.x + threadIdx.x;
    if (e >= E) return;
    float a0 = eattr[e * 2], a1 = eattr[e * 2 + 1];
    float y[32]; float s = 0.f;
#pragma unroll
    for (int o = 0; o < 32; ++o) {
        float v = b[o] + a0 * w[o * 2] + a1 * w[o * 2 + 1];
        y[o] = v; s += v;
    }
    float mu = s * (1.f / 32.f), var = 0.f;
#pragma unroll
    for (int o = 0; o < 32; ++o) { float d = y[o] - mu; var += d * d; }
    float rstd = rsqrtf(var * (1.f / 32.f) + 1e-5f);
    int t = tgt[e];
#pragma unroll
    for (int o = 0; o < 32; ++o) {
        float v = fmaxf((y[o] - mu) * rstd * g[o] + bt[o], 0.f);
        eaA[(size_t)e * 32 + o] = v;
        atomicAdd(&eaA[(size_t)(E + t) * 32 + o], v);
    }
    atomicAdd(&cnt[t], 1.f);
}

__global__ void k_loop_div(float* __restrict__ eaA, const float* __restrict__ cnt, int E, int N)
{
    int i = blockIdx.x * blockDim.x + threadIdx.x;
    if (i >= N * 32) return;
    int n = i >> 5, c = i & 31;
    eaA[(size_t)(E + n) * 32 + c] /= fmaxf(cnt[n], 1.f);
}

// ------------------------------------------------------------------
// GATv2 pass 1: logits + running max per (tgt, head)
// ------------------------------------------------------------------
__global__ __launch_bounds__(256) void k_att_logits(
    const float* __restrict__ xl, const float* __restrict__ xr,
    const float* __restrict__ eaA, const int* __restrict__ src, const int* __restrict__ tgt,
    const float* __restrict__ We, const float* __restrict__ att,
    float* __restrict__ wbuf, unsigned* __restrict__ nmax, int E, int N)
{
    __shared__ float sWe[128 * 32];
    __shared__ float sAtt[128];
    for (int i = threadIdx.x; i < 128 * 32; i += 256) sWe[i] = We[i];
    for (int i = threadIdx.x; i < 128; i += 256) sAtt[i] = att[i];
    __syncthreads();

    int gid = blockIdx.x * blockDim.x + threadIdx.x;
    if (gid >= (E + N) * 4) return;
    int e = gid >> 2, hh = gid & 3;
    int s = (e < E) ? src[e] : (e - E);
    int t = (e < E) ? tgt[e] : (e - E);

    float er[32];
    const float* ev = eaA + (size_t)e * 32;
#pragma unroll
    for (int k = 0; k < 32; ++k) er[k] = ev[k];

    const float* xs = xl + (size_t)s * 128 + hh * 32;
    const float* xt = xr + (size_t)t * 128 + hh * 32;
    float logit = 0.f;
    for (int c = 0; c < 32; ++c) {
        const float* wr = &sWe[(hh * 32 + c) * 32];
        float ee = 0.f;
#pragma unroll
        for (int k = 0; k < 32; ++k) ee += er[k] * wr[k];
        float m = xs[c] + xt[c] + ee;
        m = (m > 0.f) ? m : 0.2f * m;
        logit += m * sAtt[hh * 32 + c];
    }
    wbuf[gid] = logit;
    atomicMax(&nmax[t * 4 + hh], ford(logit));
}

// pass 2: exp(logit - max), running sum
__global__ void k_att_exp(float* __restrict__ wbuf, const unsigned* __restrict__ nmax,
                          float* __restrict__ nsum, const int* __restrict__ tgt, int E, int N)
{
    int gid = blockIdx.x * blockDim.x + threadIdx.x;
    if (gid >= (E + N) * 4) return;
    int e = gid >> 2, hh = gid & 3;
    int t = (e < E) ? tgt[e] : (e - E);
    float w = __expf(wbuf[gid] - funord(nmax[t * 4 + hh]));
    wbuf[gid] = w;
    atomicAdd(&nsum[t * 4 + hh], w);
}

// pass 3: out[tgt] += alpha * xl[src]
__global__ void k_aggregate(const float* __restrict__ wbuf, const float* __restrict__ nsum,
                            const float* __restrict__ xl, float* __restrict__ hout,
                            const int* __restrict__ src, const int* __restrict__ tgt, int E, int N)
{
    int gid = blockIdx.x * blockDim.x + threadIdx.x;
    if (gid >= (E + N) * 4) return;
    int e = gid >> 2, hh = gid & 3;
    int s = (e < E) ? src[e] : (e - E);
    int t = (e < E) ? tgt[e] : (e - E);
    float alpha = wbuf[gid] / nsum[t * 4 + hh];
    const float* xs = xl + (size_t)s * 128 + hh * 32;
    float* ho = hout + (size_t)t * 128 + hh * 32;
#pragma unroll
    for (int c = 0; c < 32; ++c) atomicAdd(&ho[c], alpha * xs[c]);
}

// node update: h = relu(LN(hout + cb + h))
__global__ void k_node_update(const float* __restrict__ hout, float* __restrict__ h,
                              const float* __restrict__ cb, const float* __restrict__ g,
                              const float* __restrict__ b, int N)
{
    int n = blockIdx.x * blockDim.x + threadIdx.x;
    if (n >= N) return;
    const float* ho = hout + (size_t)n * 128;
    float* hp = h + (size_t)n * 128;
    float s = 0.f, sq = 0.f;
    for (int c = 0; c < 128; ++c) { float v = ho[c] + cb[c] + hp[c]; s += v; sq += v * v; }
    float mu = s * (1.f / 128.f);
    float var = fmaxf(sq * (1.f / 128.f) - mu * mu, 0.f);
    float rstd = rsqrtf(var + 1e-5f);
    for (int c = 0; c < 128; ++c) {
        float v = ho[c] + cb[c] + hp[c];
        hp[c] = fmaxf((v - mu) * rstd * g[c] + b[c], 0.f);
    }
}

// ------------------------------------------------------------------
// Predicate head: fused WMMA over gathered [h[src] | h[tgt] | u] (K=258)
// One wave per 16-edge tile. relu(MLP 258->128) -> 128->9.
// ------------------------------------------------------------------
__global__ __launch_bounds__(32) void k_pred(
    const float* __restrict__ h, const int* __restrict__ src, const int* __restrict__ tgt,
    const float* __restrict__ u, const int* __restrict__ batch,
    const float* __restrict__ w1, const float* __restrict__ b1,
    const float* __restrict__ w2, const float* __restrict__ b2,
    float* __restrict__ pred)
{
    __shared__ float hid[16 * 128];
    __shared__ float ub[16 * 2];
    int lane = threadIdx.x;
    int tile = blockIdx.x;
    int row  = lane & 15;
    int e    = tile * 16 + row;
    int s = src[e], t = tgt[e];
    if (lane < 16) {
        int bb = batch[s];
        ub[row * 2 + 0] = u[bb * 2 + 0];
        ub[row * 2 + 1] = u[bb * 2 + 1];
    }
    int koff = (lane < 16) ? 0 : 2;

    v8f acc[8];
#pragma unroll
    for (int i = 0; i < 8; ++i) acc[i] = (v8f){};

    const float* hs = h + (size_t)s * 128 + koff;
    for (int k0 = 0; k0 < 128; k0 += 4) {
        v2f a; a.x = hs[k0]; a.y = hs[k0 + 1];
#pragma unroll
        for (int nt = 0; nt < 8; ++nt) {
            int col = nt * 16 + row;
            v2f bb;
            bb.x = w1[(size_t)col * 258 + k0 + koff];
            bb.y = w1[(size_t)col * 258 + k0 + koff + 1];
            acc[nt] = __builtin_amdgcn_wmma_f32_16x16x4_f32(
                false, a, false, bb, (short)0, acc[nt], false, false);
        }
    }
    const float* ht = h + (size_t)t * 128 + koff;
    for (int k0 = 0; k0 < 128; k0 += 4) {
        v2f a; a.x = ht[k0]; a.y = ht[k0 + 1];
#pragma unroll
        for (int nt = 0; nt < 8; ++nt) {
            int col = nt * 16 + row;
            v2f bb;
            bb.x = w1[(size_t)col * 258 + 128 + k0 + koff];
            bb.y = w1[(size_t)col * 258 + 128 + k0 + koff + 1];
            acc[nt] = __builtin_amdgcn_wmma_f32_16x16x4_f32(
                false, a, false, bb, (short)0, acc[nt], false, false);
        }
    }
    int rbase = (lane >> 4) * 8;
#pragma unroll
    for (int nt = 0; nt < 8; ++nt)
#pragma unroll
        for (int r = 0; r < 8; ++r)
            hid[(rbase + r) * 128 + nt * 16 + row] = acc[nt][r];
    __syncthreads();

    for (int idx = lane; idx < 16 * 128; idx += 32) {
        int rr = idx >> 7, col = idx & 127;
        float v = hid[idx] + b1[col]
                + ub[rr * 2 + 0] * w1[(size_t)col * 258 + 256]
                + ub[rr * 2 + 1] * w1[(size_t)col * 258 + 257];
        hid[idx] = fmaxf(v, 0.f);
    }
    __syncthreads();

    for (int o = lane; o < 16 * 9; o += 32) {
        int rr = o / 9, p = o % 9;
        float a = b2[p];
        for (int k = 0; k < 128; ++k) a += hid[rr * 128 + k] * w2[p * 128 + k];
        pred[(size_t)(tile * 16 + rr) * 9 + p] = a;
    }
}

// ------------------------------------------------------------------
// Pooling + graph head
// ------------------------------------------------------------------
__global__ void k_pool(const float* __restrict__ h, const int* __restrict__ batch,
                       float* __restrict__ msum, unsigned* __restrict__ mmax,
                       float* __restrict__ nb, int N)
{
    int i = blockIdx.x * blockDim.x + threadIdx.x;
    if (i >= N * 128) return;
    int n = i >> 7, c = i & 127;
    int bb = batch[n];
    float v = h[i];
    atomicAdd(&msum[bb * 128 + c], v);
    atomicMax(&mmax[bb * 128 + c], ford(v));
    if (c == 0) atomicAdd(&nb[bb], 1.f);
}

__global__ __launch_bounds__(256) void k_ghead(
    const float* __restrict__ msum, const unsigned* __restrict__ mmax,
    const float* __restrict__ nb,
    const float* __restrict__ w1, const float* __restrict__ b1,
    const float* __restrict__ w2, const float* __restrict__ b2,
    float* __restrict__ ge)
{
    __shared__ float g[16 * 256];
    __shared__ float hd[16 * 128];
    for (int i = threadIdx.x; i < 16 * 256; i += 256) {
        int bb = i >> 8, c = i & 255;
        g[i] = (c < 128) ? msum[bb * 128 + c] / fmaxf(nb[bb], 1.f)
                         : funord(mmax[bb * 128 + c - 128]);
    }
    __syncthreads();
    for (int o = threadIdx.x; o < 16 * 128; o += 256) {
        int bb = o >> 7, oc = o & 127;
        float a = b1[oc];
        for (int k = 0; k < 256; ++k) a += g[bb * 256 + k] * w1[oc * 256 + k];
        hd[o] = fmaxf(a, 0.f);
    }
    __syncthreads();
    for (int o = threadIdx.x; o < 16 * 128; o += 256) {
        int bb = o >> 7, oc = o & 127;
        float a = b2[oc];
        for (int k = 0; k < 128; ++k) a += hd[bb * 128 + k] * w2[oc * 128 + k];
        ge[o] = a;
    }
}

// ------------------------------------------------------------------
extern "C" void kernel_launch(void* const* d_in, const int* in_sizes, int n_in,
                              void* d_out, int out_size, void* d_ws, size_t ws_size,
                              hipStream_t stream)
{
    const float* x         = (const float*)d_in[0];
    const float* edge_attr = (const float*)d_in[1];
    const float* u         = (const float*)d_in[2];
    const int*   edge_idx  = (const int*)d_in[3];
    const int*   ntypes    = (const int*)d_in[4];
    const int*   batch     = (const int*)d_in[5];
    const float* type_emb  = (const float*)d_in[6];
    const float* ne_w1 = (const float*)d_in[7];
    const float* ne_b1 = (const float*)d_in[8];
    const float* ne_g1 = (const float*)d_in[9];
    const float* ne_bt1= (const float*)d_in[10];
    const float* ne_w2 = (const float*)d_in[11];
    const float* ne_b2 = (const float*)d_in[12];
    const float* ee_w  = (const float*)d_in[13];
    const float* ee_b  = (const float*)d_in[14];
    const float* ee_g  = (const float*)d_in[15];
    const float* ee_bt = (const float*)d_in[16];
    const float* Wl    = (const float*)d_in[17];
    const float* bl    = (const float*)d_in[18];
    const float* Wr    = (const float*)d_in[19];
    const float* br    = (const float*)d_in[20];
    const float* We    = (const float*)d_in[21];
    const float* att   = (const float*)d_in[22];
    const float* cb    = (const float*)d_in[23];
    const float* ln_g  = (const float*)d_in[24];
    const float* ln_b  = (const float*)d_in[25];
    const float* ph_w1 = (const float*)d_in[26];
    const float* ph_b1 = (const float*)d_in[27];
    const float* ph_w2 = (const float*)d_in[28];
    const float* ph_b2 = (const float*)d_in[29];
    const float* gh_w1 = (const float*)d_in[30];
    const float* gh_b1 = (const float*)d_in[31];
    const float* gh_w2 = (const float*)d_in[32];
    const float* gh_b2 = (const float*)d_in[33];

    const int* srcI = edge_idx;
    const int* tgtI = edge_idx + NE;

    // ---- workspace layout (floats) ----
    float* ws   = (float*)d_ws;
    float* hB   = ws;                              // NN*128
    float* t64  = hB  + (size_t)NN * 128;          // NN*64
    float* eaA  = t64 + (size_t)NN * 64;           // EA*32
    float* xlB  = eaA + (size_t)EA * 32;           // NN*128
    float* xrB  = xlB + (size_t)NN * 128;          // NN*128
    float* hout = xrB + (size_t)NN * 128;          // NN*128
    float* wbuf = hout + (size_t)NN * 128;         // EA*4
    unsigned* nmax = (unsigned*)(wbuf + (size_t)EA * 4);   // NN*4
    float* nsum = (float*)(nmax + (size_t)NN * 4); // NN*4
    float* cnt  = nsum + (size_t)NN * 4;           // NN
    float* msum = cnt + NN;                        // BG*128
    unsigned* mmax = (unsigned*)(msum + BG * 128); // BG*128
    float* nb   = (float*)(mmax + BG * 128);       // BG

    float* out_h    = (float*)d_out;
    float* out_pred = out_h + (size_t)NN * 128;
    float* out_ge   = out_pred + (size_t)NE * 9;

    const int TB = 256;

    // ---- node encoder ----
    k_node_enc1<<<CDIV(NN, TB), TB, 0, stream>>>(x, ne_w1, ne_b1, ne_g1, ne_bt1, t64, NN);
    k_gemm<6><<<CDIV(NN / 16, 8), TB, 0, stream>>>(t64, ne_w2, ne_b2, hB,
                                                   NN, 64, 64, 64, 128, 0, 0);
    k_type_concat<<<CDIV(NN * 32, TB), TB, 0, stream>>>(type_emb, ntypes, hB, NN);

    // ---- edge encoder + self-loop mean attrs ----
    k_fill_f32<<<CDIV(NN * 32, TB), TB, 0, stream>>>(eaA + (size_t)NE * 32, 0.f, NN * 32);
    k_fill_f32<<<CDIV(NN, TB), TB, 0, stream>>>(cnt, 0.f, NN);
    k_edge_enc<<<CDIV(NE, TB), TB, 0, stream>>>(edge_attr, ee_w, ee_b, ee_g, ee_bt,
                                                tgtI, eaA, cnt, NE);
    k_loop_div<<<CDIV(NN * 32, TB), TB, 0, stream>>>(eaA, cnt, NE, NN);

    // ---- GATv2 layers ----
    int eg = CDIV(EA * 4, TB);
    for (int l = 0; l < 3; ++l) {
        k_gemm<8><<<CDIV(NN / 16, 8), TB, 0, stream>>>(hB, Wl + (size_t)l * 128 * 128,
            bl + l * 128, xlB, NN, 128, 128, 128, 128, 0, 0);
        k_gemm<8><<<CDIV(NN / 16, 8), TB, 0, stream>>>(hB, Wr + (size_t)l * 128 * 128,
            br + l * 128, xrB, NN, 128, 128, 128, 128, 0, 0);
        k_fill_u32<<<CDIV(NN * 4, TB), TB, 0, stream>>>(nmax, ORD_NEG_INF, NN * 4);
        k_fill_f32<<<CDIV(NN * 4, TB), TB, 0, stream>>>(nsum, 0.f, NN * 4);
        k_fill_f32<<<CDIV(NN * 128, TB), TB, 0, stream>>>(hout, 0.f, NN * 128);
        k_att_logits<<<eg, TB, 0, stream>>>(xlB, xrB, eaA, srcI, tgtI,
            We + (size_t)l * 128 * 32, att + l * 128, wbuf, nmax, NE, NN);
        k_att_exp<<<eg, TB, 0, stream>>>(wbuf, nmax, nsum, tgtI, NE, NN);
        k_aggregate<<<eg, TB, 0, stream>>>(wbuf, nsum, xlB, hout, srcI, tgtI, NE, NN);
        k_node_update<<<CDIV(NN, TB), TB, 0, stream>>>(hout, hB, cb + l * 128,
            ln_g + l * 128, ln_b + l * 128, NN);
    }

    // ---- output h ----
    k_copy<<<CDIV(NN * 128, TB), TB, 0, stream>>>(hB, out_h, NN * 128);

    // ---- predicate head (WMMA fused) ----
    k_pred<<<NE / 16, 32, 0, stream>>>(hB, srcI, tgtI, u, batch,
                                       ph_w1, ph_b1, ph_w2, ph_b2, out_pred);

    // ---- pooling + graph head ----
    k_fill_f32<<<CDIV(BG * 128, TB), TB, 0, stream>>>(msum, 0.f, BG * 128);
    k_fill_u32<<<CDIV(BG * 128, TB), TB, 0, stream>>>(mmax, ORD_NEG_INF, BG * 128);
    k_fill_f32<<<1, TB, 0, stream>>>(nb, 0.f, BG);
    k_pool<<<CDIV(NN * 128, TB), TB, 0, stream>>>(hB, batch, msum, mmax, nb, NN);
    k_ghead<<<1, TB, 0, stream>>>(msum, mmax, nb, gh_w1, gh_b1, gh_w2, gh_b2, out_ge);

    (void)in_sizes; (void)n_in; (void)out_size; (void)ws_size;
}